// MaskedTargetPredictor_67886253081098
// MI455X (gfx1250) — compile-verified
//
#include <hip/hip_runtime.h>
#include <hip/hip_bf16.h>
#include <math.h>

// Problem constants (match reference)
#define B_      2
#define CTX_    1024
#define TGT_    256
#define DM      512          // D_MODEL
#define L_      (CTX_ + TGT_)
#define DI      1024         // D_INNER
#define DSTATE  16
#define DCONV   4
#define DTRANK  32
#define NL      2
#define LN_EPS  1e-5f

typedef __attribute__((ext_vector_type(2))) float v2f;
typedef __attribute__((ext_vector_type(8))) float v8f;

__device__ __forceinline__ float siluf(float x)     { return x / (1.f + __expf(-x)); }
__device__ __forceinline__ float softplusf(float x) { return (x > 20.f) ? x : log1pf(__expf(x)); }

#define GEMM_ACC       1   // accumulate into existing C (residual)
#define GEMM_SOFTPLUS  2   // apply softplus after bias at store

// ---------------------------------------------------------------------------
// Generic fp32 WMMA GEMM: C[M,N] (+)= A[M,K] * W[N,K]^T (+ bias) (postop)
// One wave -> one 16x64 strip (4 independent 16x16 accumulators).
// A fragment is loaded once per k-step and reused by 4 V_WMMA_F32_16X16X4_F32,
// which also keeps the 4 WMMAs RAW-independent (no D->C serialization).
// Requires: M % 16 == 0, N % 64 == 0, K % 4 == 0.
// ---------------------------------------------------------------------------
__global__ void wmma_gemm_f32(const float* __restrict__ A, int lda,
                              const float* __restrict__ W,   // N x K row-major
                              float* __restrict__ C, int ldc,
                              const float* __restrict__ bias,
                              int M, int N, int K, int flags)
{
    const int wave    = (int)((blockIdx.x * blockDim.x + threadIdx.x) >> 5);
    const int lane    = threadIdx.x & 31;
    const int nstrips = N >> 6;                // strips of 64 columns
    const int mt      = wave / nstrips;
    const int ns      = wave - mt * nstrips;
    if (mt * 16 >= M) return;                  // wave-uniform: EXEC stays all-1s

    const int lo = lane & 15;
    const int hi = lane >> 4;                  // 0: K{0,1}/M{0..7}, 1: K{2,3}/M{8..15}

    const float* arow = A + (size_t)(mt * 16 + lo) * lda;          // A row (M = lo)
    const float* wrow0 = W + (size_t)(ns * 64 +  0 + lo) * K;      // W is [N][K]
    const float* wrow1 = W + (size_t)(ns * 64 + 16 + lo) * K;
    const float* wrow2 = W + (size_t)(ns * 64 + 32 + lo) * K;
    const float* wrow3 = W + (size_t)(ns * 64 + 48 + lo) * K;

    v8f c0 = {}, c1 = {}, c2 = {}, c3 = {};
    if (flags & GEMM_ACC) {
        #pragma unroll
        for (int j = 0; j < 8; ++j) {
            const float* crow = C + (size_t)(mt * 16 + hi * 8 + j) * ldc + ns * 64 + lo;
            c0[j] = crow[0];
            c1[j] = crow[16];
            c2[j] = crow[32];
            c3[j] = crow[48];
        }
    }

    for (int kb = 0; kb < K; kb += 4) {
        const int ko = kb + hi * 2;
        v2f a, b0, b1, b2, b3;
        a[0]  = arow[ko];      a[1]  = arow[ko + 1];
        b0[0] = wrow0[ko];     b0[1] = wrow0[ko + 1];
        b1[0] = wrow1[ko];     b1[1] = wrow1[ko + 1];
        b2[0] = wrow2[ko];     b2[1] = wrow2[ko + 1];
        b3[0] = wrow3[ko];     b3[1] = wrow3[ko + 1];
        c0 = __builtin_amdgcn_wmma_f32_16x16x4_f32(false, a, false, b0, (short)0, c0, false, false);
        c1 = __builtin_amdgcn_wmma_f32_16x16x4_f32(false, a, false, b1, (short)0, c1, false, false);
        c2 = __builtin_amdgcn_wmma_f32_16x16x4_f32(false, a, false, b2, (short)0, c2, false, false);
        c3 = __builtin_amdgcn_wmma_f32_16x16x4_f32(false, a, false, b3, (short)0, c3, false, false);
    }

    const int nb = ns * 64 + lo;
    float bv0 = 0.f, bv1 = 0.f, bv2 = 0.f, bv3 = 0.f;
    if (bias) {
        bv0 = bias[nb +  0];
        bv1 = bias[nb + 16];
        bv2 = bias[nb + 32];
        bv3 = bias[nb + 48];
    }
    #pragma unroll
    for (int j = 0; j < 8; ++j) {
        float* crow = C + (size_t)(mt * 16 + hi * 8 + j) * ldc + nb;
        float v0 = c0[j] + bv0;
        float v1 = c1[j] + bv1;
        float v2 = c2[j] + bv2;
        float v3 = c3[j] + bv3;
        if (flags & GEMM_SOFTPLUS) {
            v0 = softplusf(v0); v1 = softplusf(v1);
            v2 = softplusf(v2); v3 = softplusf(v3);
        }
        crow[0]  = v0;
        crow[16] = v1;
        crow[32] = v2;
        crow[48] = v3;
    }
}

// ---------------------------------------------------------------------------
// x[b, 0:CTX, :] = context_repr
// ---------------------------------------------------------------------------
__global__ void copy_context_kernel(const float* __restrict__ ctx, float* __restrict__ x)
{
    size_t idx = (size_t)blockIdx.x * blockDim.x + threadIdx.x;
    const size_t per_b = (size_t)CTX_ * DM;
    if (idx >= (size_t)B_ * per_b) return;
    size_t b = idx / per_b, r = idx - b * per_b;
    x[b * (size_t)L_ * DM + r] = ctx[idx];
}

// ---------------------------------------------------------------------------
// tc[b,l,:] = concat(time[b,l,:], var[b,l,:])  -> (B, TGT, 2*DM)
// ---------------------------------------------------------------------------
__global__ void concat_tc_kernel(const float* __restrict__ t_time,
                                 const float* __restrict__ t_var,
                                 float* __restrict__ tc)
{
    size_t idx = (size_t)blockIdx.x * blockDim.x + threadIdx.x;
    const size_t total = (size_t)B_ * TGT_ * (2 * DM);
    if (idx >= total) return;
    size_t row = idx / (2 * DM);
    int    k   = (int)(idx - row * (2 * DM));
    tc[idx] = (k < DM) ? t_time[row * DM + k] : t_var[row * DM + (k - DM)];
}

// ---------------------------------------------------------------------------
// u = silu(causal depthwise conv(xc) + conv_b); xc = xz[..., :DI]
// ---------------------------------------------------------------------------
__global__ void conv_silu_kernel(const float* __restrict__ xz,
                                 const float* __restrict__ cw,   // DI x DCONV
                                 const float* __restrict__ cb,   // DI
                                 float* __restrict__ u)
{
    size_t idx = (size_t)blockIdx.x * blockDim.x + threadIdx.x;
    const size_t total = (size_t)B_ * L_ * DI;
    if (idx >= total) return;
    int c = (int)(idx % DI);
    size_t row = idx / DI;             // b*L + l
    int l = (int)(row % L_);
    size_t base = (row - l) * (size_t)(2 * DI);  // (b*L)*2*DI
    float acc = cb[c];
    #pragma unroll
    for (int k = 0; k < DCONV; ++k) {
        int ll = l + k - (DCONV - 1);
        if (ll >= 0)
            acc += xz[base + (size_t)ll * (2 * DI) + c] * cw[c * DCONV + k];
    }
    u[idx] = siluf(acc);
}

// ---------------------------------------------------------------------------
// Selective scan: one thread per (b, d); 16 states in registers.
// h[s] = exp(delta*A[d,s])*h[s] + delta*Bm[l,s]*u ; y = sum_s h[s]*Cm[l,s]
// ---------------------------------------------------------------------------
__global__ void scan_kernel(const float* __restrict__ delta,
                            const float* __restrict__ u,
                            const float* __restrict__ xdbl,   // (B*L, 64): [dt|Bm|Cm]
                            const float* __restrict__ A_log,  // DI x DSTATE
                            float* __restrict__ y)
{
    int t = blockIdx.x * blockDim.x + threadIdx.x;
    if (t >= B_ * DI) return;
    int b = t / DI, d = t - b * DI;

    float Arow[DSTATE], h[DSTATE];
    #pragma unroll
    for (int s = 0; s < DSTATE; ++s) {
        Arow[s] = -__expf(A_log[d * DSTATE + s]);
        h[s] = 0.f;
    }

    const float* dp = delta + (size_t)b * L_ * DI + d;
    const float* up = u     + (size_t)b * L_ * DI + d;
    const float* xd = xdbl  + (size_t)b * L_ * 64;
    float*       yp = y     + (size_t)b * L_ * DI + d;

    for (int l = 0; l < L_; ++l) {
        float de = dp[(size_t)l * DI];
        float uu = up[(size_t)l * DI];
        const float* Bp = xd + (size_t)l * 64 + DTRANK;
        const float* Cp = Bp + DSTATE;
        float dub = de * uu;
        float acc = 0.f;
        #pragma unroll
        for (int s = 0; s < DSTATE; ++s) {
            float dA = __expf(de * Arow[s]);
            h[s] = dA * h[s] + dub * Bp[s];
            acc += h[s] * Cp[s];
        }
        yp[(size_t)l * DI] = acc;
    }
}

// ---------------------------------------------------------------------------
// y = (y + u * D_skip[c]) * silu(z),  z = xz[..., DI + c]
// ---------------------------------------------------------------------------
__global__ void gate_kernel(float* __restrict__ y,
                            const float* __restrict__ u,
                            const float* __restrict__ xz,
                            const float* __restrict__ dskip)
{
    size_t idx = (size_t)blockIdx.x * blockDim.x + threadIdx.x;
    const size_t total = (size_t)B_ * L_ * DI;
    if (idx >= total) return;
    int c = (int)(idx % DI);
    size_t row = idx / DI;
    float z = xz[row * (size_t)(2 * DI) + DI + c];
    y[idx] = (y[idx] + u[idx] * dskip[c]) * siluf(z);
}

// ---------------------------------------------------------------------------
// LayerNorm over DM=512 per row; one wave32 per row (16 elems/lane).
// ---------------------------------------------------------------------------
__global__ void ln_kernel(float* __restrict__ x,
                          const float* __restrict__ g,
                          const float* __restrict__ bb, int rows)
{
    int wave = (int)((blockIdx.x * blockDim.x + threadIdx.x) >> 5);
    int lane = threadIdx.x & 31;
    if (wave >= rows) return;
    float* r = x + (size_t)wave * DM;

    float v[16];
    float s = 0.f;
    #pragma unroll
    for (int i = 0; i < 16; ++i) { v[i] = r[lane + i * 32]; s += v[i]; }
    #pragma unroll
    for (int o = 16; o > 0; o >>= 1) s += __shfl_xor(s, o, 32);
    float mu = s * (1.f / DM);

    float q = 0.f;
    #pragma unroll
    for (int i = 0; i < 16; ++i) { float d = v[i] - mu; q += d * d; }
    #pragma unroll
    for (int o = 16; o > 0; o >>= 1) q += __shfl_xor(q, o, 32);
    float rs = rsqrtf(q * (1.f / DM) + LN_EPS);

    #pragma unroll
    for (int i = 0; i < 16; ++i) {
        int col = lane + i * 32;
        r[col] = (v[i] - mu) * rs * g[col] + bb[col];
    }
}

// ---------------------------------------------------------------------------
extern "C" void kernel_launch(void* const* d_in, const int* in_sizes, int n_in,
                              void* d_out, int out_size, void* d_ws, size_t ws_size,
                              hipStream_t stream)
{
    (void)in_sizes; (void)n_in; (void)out_size; (void)ws_size;

    const float* context = (const float*)d_in[0];
    const float* t_time  = (const float*)d_in[1];
    const float* t_var   = (const float*)d_in[2];
    // d_in[3] = target_count (int scalar) — compile-time constant TGT_
    const float* pos_w   = (const float*)d_in[4];
    const float* pos_b   = (const float*)d_in[5];
    const float* outp_w  = (const float*)d_in[6];
    const float* outp_b  = (const float*)d_in[7];
    const float* ln_g    = (const float*)d_in[8];
    const float* ln_b    = (const float*)d_in[9];
    const float* in_w    = (const float*)d_in[10];
    const float* conv_w  = (const float*)d_in[11];
    const float* conv_b  = (const float*)d_in[12];
    const float* xproj_w = (const float*)d_in[13];
    const float* dt_w    = (const float*)d_in[14];
    const float* dt_b    = (const float*)d_in[15];
    const float* A_log   = (const float*)d_in[16];
    const float* D_skip  = (const float*)d_in[17];
    const float* out_w   = (const float*)d_in[18];
    float* out = (float*)d_out;

    // Workspace carve (floats)
    float* ws = (float*)d_ws;
    size_t off = 0;
    float* x     = ws + off; off += (size_t)B_ * L_   * DM;        // residual stream
    float* tc    = ws + off; off += (size_t)B_ * TGT_ * (2 * DM);  // [time|var]
    float* xz    = ws + off; off += (size_t)B_ * L_   * (2 * DI);  // in-proj out
    float* u     = ws + off; off += (size_t)B_ * L_   * DI;        // conv+silu out
    float* xdbl  = ws + off; off += (size_t)B_ * L_   * 64;        // [dt|B|C]
    float* delta = ws + off; off += (size_t)B_ * L_   * DI;        // softplus(dtproj)
    float* y     = ws + off; off += (size_t)B_ * L_   * DI;        // scan out / gated

    auto gemm = [&](const float* A, int lda, const float* W, float* C, int ldc,
                    const float* bias, int M, int N, int K, int flags) {
        int waves  = (M / 16) * (N / 64);   // 16x64 strip per wave
        int blocks = (waves * 32 + 255) / 256;
        wmma_gemm_f32<<<blocks, 256, 0, stream>>>(A, lda, W, C, ldc, bias, M, N, K, flags);
    };

    const int M_all = B_ * L_;   // 2560 rows through the mamba stack

    // 1) residual stream init: context rows + positional projection of targets
    {
        size_t n = (size_t)B_ * CTX_ * DM;
        copy_context_kernel<<<(int)((n + 255) / 256), 256, 0, stream>>>(context, x);
        n = (size_t)B_ * TGT_ * (2 * DM);
        concat_tc_kernel<<<(int)((n + 255) / 256), 256, 0, stream>>>(t_time, t_var, tc);
        for (int b = 0; b < B_; ++b)
            gemm(tc + (size_t)b * TGT_ * (2 * DM), 2 * DM, pos_w,
                 x + ((size_t)b * L_ + CTX_) * DM, DM, pos_b,
                 TGT_, DM, 2 * DM, 0);
    }

    // 2) Mamba layers
    for (int layer = 0; layer < NL; ++layer) {
        const float* l_in_w    = in_w    + (size_t)layer * (2 * DI) * DM;
        const float* l_conv_w  = conv_w  + (size_t)layer * DI * DCONV;
        const float* l_conv_b  = conv_b  + (size_t)layer * DI;
        const float* l_xproj_w = xproj_w + (size_t)layer * 64 * DI;
        const float* l_dt_w    = dt_w    + (size_t)layer * DI * DTRANK;
        const float* l_dt_b    = dt_b    + (size_t)layer * DI;
        const float* l_A_log   = A_log   + (size_t)layer * DI * DSTATE;
        const float* l_D_skip  = D_skip  + (size_t)layer * DI;
        const float* l_out_w   = out_w   + (size_t)layer * DM * DI;

        // in-projection: xz = x @ in_w^T      (2560 x 2048 x 512)
        gemm(x, DM, l_in_w, xz, 2 * DI, nullptr, M_all, 2 * DI, DM, 0);

        // depthwise causal conv + SiLU -> u
        {
            size_t n = (size_t)B_ * L_ * DI;
            conv_silu_kernel<<<(int)((n + 255) / 256), 256, 0, stream>>>(
                xz, l_conv_w, l_conv_b, u);
        }

        // x-projection: xdbl = u @ xproj_w^T  (2560 x 64 x 1024)
        gemm(u, DI, l_xproj_w, xdbl, 64, nullptr, M_all, 64, DI, 0);

        // delta = softplus(dt @ dt_w^T + dt_b)  (2560 x 1024 x 32, lda=64)
        gemm(xdbl, 64, l_dt_w, delta, DI, l_dt_b, M_all, DI, DTRANK, GEMM_SOFTPLUS);

        // selective scan -> y
        scan_kernel<<<(B_ * DI + 255) / 256, 256, 0, stream>>>(delta, u, xdbl, l_A_log, y);

        // gating: y = (y + u*D) * silu(z)
        {
            size_t n = (size_t)B_ * L_ * DI;
            gate_kernel<<<(int)((n + 255) / 256), 256, 0, stream>>>(y, u, xz, l_D_skip);
        }

        // out-projection with fused residual: x += y @ out_w^T  (2560 x 512 x 1024)
        gemm(y, DI, l_out_w, x, DM, nullptr, M_all, DM, DI, GEMM_ACC);
    }

    // 3) LayerNorm (wave per row)
    ln_kernel<<<(M_all * 32 + 255) / 256, 256, 0, stream>>>(x, ln_g, ln_b, M_all);

    // 4) output projection on the last TGT rows of each batch
    for (int b = 0; b < B_; ++b)
        gemm(x + ((size_t)b * L_ + CTX_) * DM, DM, outp_w,
             out + (size_t)b * TGT_ * DM, DM, outp_b,
             TGT_, DM, DM, 0);
}